// StructuredPerceptron_37417755083582
// MI455X (gfx1250) — compile-verified
//
#include <hip/hip_runtime.h>

// Problem constants from the reference: (B, S, T) = (64, 2048, 512)
#define BATCH 64
#define SEQ   2048
#define NTAG  512
#define CHUNKS 8          // SEQ / 256 positions handled per block

typedef float v2f __attribute__((ext_vector_type(2)));
typedef float v8f __attribute__((ext_vector_type(8)));

// Exact f32 wave32 reduction on the matrix pipe:
// A (16x4) holds the 32 lane partials (a.x = partial, a.y = 0), B = ones.
// D[m,n] = sum_k A[m,k] = partial(lane m) + partial(lane m+16).
// Lanes 0-15 hold D[0..7,n] in c[0..7]; lanes 16-31 hold D[8..15,n].
// Sum the 8 accumulators, then fold halves with one shfl_xor(16).
__device__ __forceinline__ float wave_sum_wmma(float x) {
  v2f a;    a.x = x;    a.y = 0.0f;
  v2f ones; ones.x = 1.0f; ones.y = 1.0f;
  v8f c = {};
  c = __builtin_amdgcn_wmma_f32_16x16x4_f32(
      /*neg_a=*/false, a, /*neg_b=*/false, ones,
      /*c_mod=*/(short)0, c, /*reuse_a=*/false, /*reuse_b=*/false);
  float t = c[0] + c[1] + c[2] + c[3] + c[4] + c[5] + c[6] + c[7];
  t += __shfl_xor(t, 16, 32);
  return t;  // full 32-lane sum, valid in every lane
}

__global__ void sp_zero_ws(float* __restrict__ ws) {
  ws[threadIdx.x] = 0.0f;  // 64 per-batch accumulators
}

// One (b, s) pair per thread. 512 blocks x 256 threads.
__global__ void sp_accum(const float* __restrict__ unary,
                         const float* __restrict__ binary,
                         const int*   __restrict__ tags,
                         const int*   __restrict__ pred,
                         float*       __restrict__ ws) {
  const int b = (int)(blockIdx.x >> 3);              // blockIdx / CHUNKS
  const int s = (int)((blockIdx.x & 7) << 8) + (int)threadIdx.x;
  const long long idx = (long long)b * SEQ + s;

  const int tg = tags[idx];
  const int tp = pred[idx];
  const float* __restrict__ row = unary + idx * (long long)NTAG;

  // diff contribution = predicted score - gold score (unary part)
  float acc = row[tp] - row[tg];

  // transition part: binary[tag_s, tag_{s+1}], valid for s < SEQ-1
  if (s + 1 < SEQ) {
    const int tg2 = tags[idx + 1];
    const int tp2 = pred[idx + 1];
    acc += binary[tp * NTAG + tp2] - binary[tg * NTAG + tg2];
  }

  // Block reduction: WMMA wave sums -> LDS -> one atomic per block.
  const float wsum = wave_sum_wmma(acc);           // EXEC all-1s: full block active
  __shared__ float lds[8];
  const int lane = (int)(threadIdx.x & 31);
  const int wid  = (int)(threadIdx.x >> 5);
  if (lane == 0) lds[wid] = wsum;
  __syncthreads();
  if (threadIdx.x == 0) {
    float tot = 0.0f;
#pragma unroll
    for (int i = 0; i < 8; ++i) tot += lds[i];
    atomicAdd(&ws[b], tot);                        // global_atomic_add_f32
  }
}

// 64 threads (2 full waves): ReLU per batch, reduce, write scalar.
__global__ void sp_finalize(const float* __restrict__ ws,
                            float*       __restrict__ out) {
  const float v = fmaxf(ws[threadIdx.x], 0.0f);
  const float wsum = wave_sum_wmma(v);
  __shared__ float lds[2];
  if ((threadIdx.x & 31) == 0) lds[threadIdx.x >> 5] = wsum;
  __syncthreads();
  if (threadIdx.x == 0) out[0] = lds[0] + lds[1];
}

extern "C" void kernel_launch(void* const* d_in, const int* in_sizes, int n_in,
                              void* d_out, int out_size, void* d_ws, size_t ws_size,
                              hipStream_t stream) {
  const float* unary  = (const float*)d_in[0];  // (64, 2048, 512) f32
  const float* binary = (const float*)d_in[1];  // (512, 512) f32
  const int*   tags   = (const int*)d_in[2];    // (64, 2048) i32
  const int*   pred   = (const int*)d_in[3];    // (64, 2048) i32
  // d_in[4] = mask: unused by the reference computation

  float* ws  = (float*)d_ws;    // first 64 floats: per-batch diff accumulators
  float* out = (float*)d_out;   // single f32 loss

  sp_zero_ws<<<1, BATCH, 0, stream>>>(ws);
  sp_accum <<<BATCH * CHUNKS, 256, 0, stream>>>(unary, binary, tags, pred, ws);
  sp_finalize<<<1, BATCH, 0, stream>>>(ws, out);
}